// AttentionModule_6459630813574
// MI455X (gfx1250) — compile-verified
//
#include <hip/hip_runtime.h>

// ---------------------------------------------------------------------------
// MI455X (gfx1250) fused attention-pool kernel.
//
// Math: out = ((alpha @ X) / sum(alpha)) @ W_sum,  alpha_j = exp(s0 + X[j].b_w)
// The constant factor exp(s0) (s0 = a_w.X[0]) cancels between numerator and
// denominator, so a_alpha is provably irrelevant -> single streaming pass.
//
// Roofline: 102.4 MB of X @ 23.3 TB/s -> ~4.4 us floor; ~0.1 GFLOP total.
// Pure bandwidth problem: one coalesced pass, LDS-staged 16-row tiles,
// WMMA f32 16x16x4 for the per-tile GEMV, deterministic 2-stage reduction.
// ---------------------------------------------------------------------------

typedef __attribute__((ext_vector_type(2))) float v2f;
typedef __attribute__((ext_vector_type(8))) float v8f;

#define DDIM 128
#define TILE_ROWS 16
#define WAVES_PER_BLOCK 4
#define BLOCK_THREADS (WAVES_PER_BLOCK * 32)
#define MAX_BLOCKS 500

// ---------------------------------------------------------------- phase 0 --
// b_w[d] = sum_k b_alpha[k] * W_alpha[k, d]          (1 block, 128 threads)
__global__ void bw_kernel(const float* __restrict__ b_alpha,
                          const float* __restrict__ W_alpha,
                          float* __restrict__ bw) {
    const int d = threadIdx.x;
    float s = 0.f;
    #pragma unroll 8
    for (int k = 0; k < DDIM; ++k)
        s = fmaf(b_alpha[k], W_alpha[k * DDIM + d], s);
    bw[d] = s;
}

// ---------------------------------------------------------------- phase 1 --
// Stream X in 16x128 tiles. Per wave:
//   - stage tile in wave-private LDS (global b128 -> ds b128, coalesced)
//   - dots = X_tile . b_w via 32 chained V_WMMA_F32_16X16X4_F32
//     (B = b_w chunk replicated over N, so D[m][n] = dot[m] for all n)
//   - alpha = exp(dots); broadcast 16 alphas to all lanes (half-wave shuffles)
//   - acc[4 cols/lane] += alpha_r * X_tile[r]  (64 VALU FMAs per tile)
// Then LDS block-reduce and store per-block partials (deterministic).
__global__ void __launch_bounds__(BLOCK_THREADS)
main_kernel(const float* __restrict__ X,
            const float* __restrict__ bw_g,
            float* __restrict__ partials,
            int nRows, int numTiles) {
    __shared__ float s_bw[DDIM];
    __shared__ float s_tile[WAVES_PER_BLOCK][TILE_ROWS * DDIM];

    const int tid  = threadIdx.x;
    const int lane = tid & 31;
    const int wave = tid >> 5;

    if (tid < DDIM) s_bw[tid] = bw_g[tid];
    __syncthreads();

    const int gwave  = blockIdx.x * WAVES_PER_BLOCK + wave;
    const int nwaves = gridDim.x * WAVES_PER_BLOCK;

    float4 acc = {0.f, 0.f, 0.f, 0.f};
    float  asum = 0.f;

    // WMMA A-layout (32-bit A 16x4): lane holds row (lane&15), K = 2*(lane>>4)+{0,1}
    const int mrow = lane & 15;
    const int koff = (lane >> 4) * 2;
    float* tile = &s_tile[wave][0];

    for (int t = gwave; t < numTiles; t += nwaves) {
        const long base = (long)t * (TILE_ROWS * DDIM);

        // Prefetch next tile for this wave (global_prefetch_b8).
        if (t + nwaves < numTiles)
            __builtin_prefetch(X + base + (long)nwaves * (TILE_ROWS * DDIM), 0, 0);

        // ---- cooperative, coalesced global->LDS stage: 512 float4, 16/lane
        const float4* src = (const float4*)(X + base);
        float4*       dst = (float4*)tile;
        const long rowBase = (long)t * TILE_ROWS;
        #pragma unroll
        for (int i = 0; i < 16; ++i) {
            const int v   = lane + 32 * i;   // float4 index within tile
            const int row = v >> 5;          // 32 float4 per 128-float row
            float4 val = {0.f, 0.f, 0.f, 0.f};
            if (rowBase + row < nRows) val = src[v];
            dst[v] = val;                    // tail rows zeroed
        }
        // EXEC fully restored here; WMMA below runs with all lanes active.

        // ---- dots via WMMA chain over K=128 (32 x 16x16x4 f32)
        v8f c = {0.f, 0.f, 0.f, 0.f, 0.f, 0.f, 0.f, 0.f};
        const float* rp = tile + mrow * DDIM;
        #pragma unroll
        for (int k0 = 0; k0 < DDIM; k0 += 4) {
            v2f a = *(const v2f*)(rp   + k0 + koff);   // A: X tile chunk
            v2f b = *(const v2f*)(s_bw + k0 + koff);   // B: b_w replicated over N
            c = __builtin_amdgcn_wmma_f32_16x16x4_f32(
                    /*neg_a=*/false, a, /*neg_b=*/false, b,
                    /*c_mod=*/(short)0, c, /*reuse_a=*/false, /*reuse_b=*/false);
        }
        // C/D layout: lanes 0-15 VGPR r -> row r ; lanes 16-31 VGPR r -> row 8+r.

        // ---- alpha = exp(dot); broadcast all 16 row-alphas to every lane
        float alpha[16];
        #pragma unroll
        for (int r = 0; r < 8; ++r) {
            const float e = __expf(c[r]);
            alpha[r]     = __shfl(e, 0, 32);   // rows 0..7  (from lane 0)
            alpha[r + 8] = __shfl(e, 16, 32);  // rows 8..15 (from lane 16)
        }

        // ---- weighted accumulation: 4 columns per lane, b128 LDS reads
        #pragma unroll
        for (int r = 0; r < 16; ++r) {
            if (rowBase + r >= nRows) alpha[r] = 0.f;   // kill tail rows
            asum += alpha[r];
            const float4 xr = *(const float4*)(tile + r * DDIM + lane * 4);
            acc.x = fmaf(alpha[r], xr.x, acc.x);
            acc.y = fmaf(alpha[r], xr.y, acc.y);
            acc.z = fmaf(alpha[r], xr.z, acc.z);
            acc.w = fmaf(alpha[r], xr.w, acc.w);
        }
    }

    // ---- block reduction (reuse the tile buffers; wave-private writes)
    {
        float* red = &s_tile[wave][0];
        *(float4*)(red + lane * 4) = acc;        // cols 4L..4L+3
        if (lane == 0) red[DDIM] = asum;         // asum is wave-uniform
    }
    __syncthreads();
    if (tid < DDIM) {
        float tot = 0.f;
        #pragma unroll
        for (int w = 0; w < WAVES_PER_BLOCK; ++w) tot += s_tile[w][tid];
        partials[blockIdx.x * (DDIM + 1) + tid] = tot;
    }
    if (tid == 0) {
        float at = 0.f;
        #pragma unroll
        for (int w = 0; w < WAVES_PER_BLOCK; ++w) at += s_tile[w][DDIM];
        partials[blockIdx.x * (DDIM + 1) + DDIM] = at;
    }
}

// ---------------------------------------------------------------- phase 2 --
// Deterministic partial reduce, normalize, and out = sum_output @ W_sum.
__global__ void finalize_kernel(const float* __restrict__ partials,
                                const float* __restrict__ W_sum,
                                float* __restrict__ out, int nblocks) {
    __shared__ float s_s[DDIM];
    const int t = threadIdx.x;   // 128 threads
    float accv = 0.f, asum = 0.f;
    for (int b = 0; b < nblocks; ++b) {
        accv += partials[b * (DDIM + 1) + t];
        asum += partials[b * (DDIM + 1) + DDIM];
    }
    s_s[t] = accv / asum;
    __syncthreads();
    float o = 0.f;
    #pragma unroll 8
    for (int d = 0; d < DDIM; ++d)
        o = fmaf(s_s[d], W_sum[d * DDIM + t], o);
    out[t] = o;
}

// --------------------------------------------------------------------------
extern "C" void kernel_launch(void* const* d_in, const int* in_sizes, int n_in,
                              void* d_out, int out_size, void* d_ws, size_t ws_size,
                              hipStream_t stream) {
    // setup_inputs order: X, W_sum, W_alpha, a_alpha, b_alpha (all f32)
    const float* X       = (const float*)d_in[0];
    const float* W_sum   = (const float*)d_in[1];
    const float* W_alpha = (const float*)d_in[2];
    // d_in[3] = a_alpha: exp(a_w.X[0]) is a common factor that cancels exactly.
    const float* b_alpha = (const float*)d_in[4];

    const int nRows    = in_sizes[0] / DDIM;                      // 200000
    const int numTiles = (nRows + TILE_ROWS - 1) / TILE_ROWS;     // 12500

    float* bw       = (float*)d_ws;       // 128 floats
    float* partials = bw + DDIM;          // MAX_BLOCKS * 129 floats (~258 KB)

    int blocks = MAX_BLOCKS;
    const int maxUseful = (numTiles + WAVES_PER_BLOCK - 1) / WAVES_PER_BLOCK;
    if (blocks > maxUseful) blocks = maxUseful;

    bw_kernel<<<1, DDIM, 0, stream>>>(b_alpha, W_alpha, bw);
    main_kernel<<<blocks, BLOCK_THREADS, 0, stream>>>(X, bw, partials, nRows, numTiles);
    finalize_kernel<<<1, DDIM, 0, stream>>>(partials, W_sum, (float*)d_out, blocks);
}